// GaussianAttention_45183055954091
// MI455X (gfx1250) — compile-verified
//
#include <hip/hip_runtime.h>
#include <hip/hip_bf16.h>
#include <math.h>

typedef __attribute__((ext_vector_type(2))) float v2f;
typedef __attribute__((ext_vector_type(8))) float v8f;

#define B_    4
#define H_    48
#define W_    48
#define CIN   256
#define CMID  16
#define HW    (H_ * W_)        // 2304
#define M_TOT (B_ * HW)        // 9216
#define PDF_SZ  ((size_t)B_ * HW * HW)   // 21233664
#define MEAN_SZ (M_TOT * 2)              // 18432
// workspace float offsets
#define OFF_VRAW  0
#define OFF_MEAN  (OFF_VRAW + M_TOT * 2)
#define OFF_ICOV  (OFF_MEAN + M_TOT * 2)
#define OFF_COEF  (OFF_ICOV + M_TOT * 2)
#define OFF_SUM   (OFF_COEF + M_TOT)
#define OFF_SUMSQ (OFF_SUM + B_)
#define OFF_MU    (OFF_SUMSQ + B_)
#define OFF_RSTD  (OFF_MU + B_)

// ---------------------------------------------------------------------------
// 0) zero the per-batch atomic accumulators (graph replay safe)
// ---------------------------------------------------------------------------
__global__ void ga_init(float* ws) {
    int t = threadIdx.x;
    if (t < 2 * B_) ws[OFF_SUM + t] = 0.0f;
}

// ---------------------------------------------------------------------------
// 1) tt = tanh(x @ map_w^T + map_b) via V_WMMA_F32_16X16X4_F32, then the tiny
//    16->2 projections per row + atomic per-batch sum / sumsq of v.
//    One wave (32 threads) per 16-row tile of the (9216 x 16) result.
// ---------------------------------------------------------------------------
__global__ __launch_bounds__(32)
void ga_map_wmma(const float* __restrict__ x,
                 const float* __restrict__ map_w,
                 const float* __restrict__ map_b,
                 const float* __restrict__ mean_w,
                 const float* __restrict__ mean_b,
                 const float* __restrict__ cov_w,
                 const float* __restrict__ cov_b,
                 float* ws,
                 float* __restrict__ out_mean) {
    __shared__ float tt[16 * 16];

    const int lane = threadIdx.x;     // 0..31, wave32
    const int half = lane >> 4;       // 0: K=0,1   1: K=2,3
    const int lr   = lane & 15;
    const int m0   = blockIdx.x * 16;

    // A fragment source: row (m0+lr) of x, K offsets (k+2*half, k+2*half+1)
    const float* __restrict__ xrow = x + (size_t)(m0 + lr) * CIN + 2 * half;
    // B fragment source: B[k][n] = map_w[n][k]; lane holds column n = lr
    const float* __restrict__ brow = map_w + (size_t)lr * CIN + 2 * half;

    v8f acc = {};
    #pragma unroll 4
    for (int k = 0; k < CIN; k += 4) {
        v2f a, b;
        a.x = xrow[k];
        a.y = xrow[k + 1];
        b.x = brow[k];
        b.y = brow[k + 1];
        acc = __builtin_amdgcn_wmma_f32_16x16x4_f32(
                  /*neg_a=*/false, a, /*neg_b=*/false, b,
                  /*c_mod=*/(short)0, acc, /*reuse_a=*/false, /*reuse_b=*/false);
    }

    // C/D layout: VGPR r -> (M=r, N=lane) for lanes 0-15, (M=r+8, N=lane-16)
    const float bias = map_b[lr];
    #pragma unroll
    for (int r = 0; r < 8; ++r) {
        const int m = r + 8 * half;
        tt[m * 16 + lr] = tanhf(acc[r] + bias);
    }
    __syncthreads();

    if (lane < 16) {
        const int g = m0 + lane;                 // global row 0..9215
        float mo0 = mean_b[0], mo1 = mean_b[1];
        float v0  = cov_b[0],  v1  = cov_b[1];
        #pragma unroll
        for (int j = 0; j < 16; ++j) {
            const float t = tt[lane * 16 + j];
            mo0 += t * mean_w[j];
            mo1 += t * mean_w[16 + j];
            v0  += t * cov_w[j];
            v1  += t * cov_w[16 + j];
        }
        const int b = g / HW;
        const int p = g % HW;
        const float mean0 = (float)(p % W_) + mo0;   // coord x = p % W
        const float mean1 = (float)(p / W_) + mo1;   // coord y = p / W
        ws[OFF_MEAN + 2 * g]     = mean0;
        ws[OFF_MEAN + 2 * g + 1] = mean1;
        out_mean[2 * g]     = mean0;                 // output #2: mean (b,hw,1,2)
        out_mean[2 * g + 1] = mean1;
        ws[OFF_VRAW + 2 * g]     = v0;
        ws[OFF_VRAW + 2 * g + 1] = v1;
        atomicAdd(&ws[OFF_SUM + b],   v0 + v1);
        atomicAdd(&ws[OFF_SUMSQ + b], v0 * v0 + v1 * v1);
    }
}

// ---------------------------------------------------------------------------
// 2) per-batch mean / rstd of v over (hw, 2)
// ---------------------------------------------------------------------------
__global__ void ga_stats(float* ws) {
    int t = threadIdx.x;
    if (t < B_) {
        const float n   = (float)(HW * 2);
        const float mu  = ws[OFF_SUM + t] / n;
        const float var = ws[OFF_SUMSQ + t] / n - mu * mu + 1e-5f;
        ws[OFF_MU + t]   = mu;
        ws[OFF_RSTD + t] = rsqrtf(var);
    }
}

// ---------------------------------------------------------------------------
// 3) normalize v, sigmoid*5+0.05, icov, det, pdf coefficient
// ---------------------------------------------------------------------------
__global__ __launch_bounds__(256)
void ga_final(float* ws, float* __restrict__ out_det) {
    const int g = blockIdx.x * blockDim.x + threadIdx.x;
    if (g >= M_TOT) return;
    const int b = g / HW;
    const float mu   = ws[OFF_MU + b];
    const float rstd = ws[OFF_RSTD + b];
    float v0 = (ws[OFF_VRAW + 2 * g]     - mu) * rstd;
    float v1 = (ws[OFF_VRAW + 2 * g + 1] - mu) * rstd;
    v0 = 5.0f / (1.0f + expf(-v0)) + 0.05f;
    v1 = 5.0f / (1.0f + expf(-v1)) + 0.05f;
    ws[OFF_ICOV + 2 * g]     = 1.0f / v0;
    ws[OFF_ICOV + 2 * g + 1] = 1.0f / v1;
    const float det = v0 * v1;
    out_det[g] = det;                                // output #3: det (b,hw)
    ws[OFF_COEF + g] = 3.0f / (6.28f * sqrtf(det));
}

// ---------------------------------------------------------------------------
// 4) pdf (b,hw,hw): 85 MB of stores -> pure HBM-write-bound; float4 stores.
//    One block per output row; coords computed arithmetically.
// ---------------------------------------------------------------------------
__global__ __launch_bounds__(256)
void ga_pdf(const float* __restrict__ ws, float* __restrict__ pdf) {
    const int row = blockIdx.x;                      // b*HW + i
    const float m0   = ws[OFF_MEAN + 2 * row];
    const float m1   = ws[OFF_MEAN + 2 * row + 1];
    const float ic0  = ws[OFF_ICOV + 2 * row];
    const float ic1  = ws[OFF_ICOV + 2 * row + 1];
    const float coef = ws[OFF_COEF + row];

    float4* __restrict__ outp = (float4*)(pdf + (size_t)row * HW);
    for (int q = threadIdx.x; q < HW / 4; q += 256) {
        const int j = q * 4;                         // 48 % 4 == 0: same cy for all 4
        float d0 = (float)(j % W_) - m0;
        const float d1 = (float)(j / W_) - m1;
        const float e1 = d1 * d1 * ic1;
        float4 r;
        r.x = coef * __expf(-0.5f * (d0 * d0 * ic0 + e1)); d0 += 1.0f;
        r.y = coef * __expf(-0.5f * (d0 * d0 * ic0 + e1)); d0 += 1.0f;
        r.z = coef * __expf(-0.5f * (d0 * d0 * ic0 + e1)); d0 += 1.0f;
        r.w = coef * __expf(-0.5f * (d0 * d0 * ic0 + e1));
        outp[q] = r;
    }
}

// ---------------------------------------------------------------------------
extern "C" void kernel_launch(void* const* d_in, const int* in_sizes, int n_in,
                              void* d_out, int out_size, void* d_ws, size_t ws_size,
                              hipStream_t stream) {
    const float* x      = (const float*)d_in[0];
    const float* map_w  = (const float*)d_in[1];
    const float* map_b  = (const float*)d_in[2];
    const float* mean_w = (const float*)d_in[3];
    const float* mean_b = (const float*)d_in[4];
    const float* cov_w  = (const float*)d_in[5];
    const float* cov_b  = (const float*)d_in[6];

    float* out      = (float*)d_out;
    float* pdf      = out;                   // (b, hw, hw)
    float* out_mean = out + PDF_SZ;          // (b, hw, 1, 2)
    float* out_det  = out_mean + MEAN_SZ;    // (b, hw)
    float* ws       = (float*)d_ws;

    ga_init<<<1, 32, 0, stream>>>(ws);
    ga_map_wmma<<<M_TOT / 16, 32, 0, stream>>>(x, map_w, map_b, mean_w, mean_b,
                                               cov_w, cov_b, ws, out_mean);
    ga_stats<<<1, 32, 0, stream>>>(ws);
    ga_final<<<(M_TOT + 255) / 256, 256, 0, stream>>>(ws, out_det);
    ga_pdf<<<M_TOT, 256, 0, stream>>>(ws, pdf);
}